// FDEmbeddingBlock_738734375480
// MI455X (gfx1250) — compile-verified
//
#include <hip/hip_runtime.h>
#include <hip/hip_bf16.h>
#include <math.h>

// ---------------------------------------------------------------------------
// FDEmbeddingBlock: rfft(65536) keep 4000 bins -> layernorm(8000) -> GEMM
// (2048x8000)x(8000x512)+bias -> SiLU.   Output: 2048x512 fp32.
//
// MI455X reasoning: reading x (512 MB) at 23.3 TB/s is the ~22us floor.
// FFT done per-row in LDS (256 KB working set fits CDNA5's 320KB/WGP),
// loaded via TDM (tensor_load_to_lds) where available, with a streaming
// fallback. DIF ordering keeps the global read fully coalesced. Twiddles
// come from a per-thread complex recurrence (pos steps by 512 mod half),
// so transcendental pressure is negligible. GEMM is tiny (16.8 GFLOP)
// -> fp32 WMMA (V_WMMA_F32_16X16X4_F32) for accuracy.
// ---------------------------------------------------------------------------

#define FFT_N    65536
#define FFT_NC   32768      // complex length after even/odd packing
#define KEEP     4000
#define FEAT     8000       // 2*KEEP
#define BROWS    2048
#define OUTC     512
#define T1       512        // threads in FFT kernel
#define PI_F     3.14159265358979f
#define TWO_PI_F 6.28318530717959f

typedef __attribute__((ext_vector_type(2))) float        v2f;
typedef __attribute__((ext_vector_type(8))) float        v8f;
typedef __attribute__((ext_vector_type(4))) unsigned int v4u;
typedef __attribute__((ext_vector_type(8))) int          v8i;
typedef __attribute__((ext_vector_type(4))) int          v4i;

#if defined(__AMDGCN__) && __has_builtin(__builtin_amdgcn_tensor_load_to_lds) && \
    __has_builtin(__builtin_amdgcn_s_wait_tensorcnt)
#define USE_TDM 1
#warning "CDNA5: TDM path ENABLED (tensor_load_to_lds emitted)"
#else
#define USE_TDM 0
#if defined(__AMDGCN__)
#warning "CDNA5: TDM path DISABLED (builtin missing); using vector-load fallback"
#endif
#endif

// ---------------------------------------------------------------------------
// Kernel 1: per-row real FFT (packed complex, radix-2 DIF) in LDS + layernorm.
// One block per row. Dynamic LDS: 32768 float2 (FFT) + 8000 float (features).
// ---------------------------------------------------------------------------
__global__ __launch_bounds__(T1)
void fft_ln_kernel(const float* __restrict__ x,
                   const float* __restrict__ ln_w,
                   const float* __restrict__ ln_b,
                   float* __restrict__ y /* BROWS x FEAT */)
{
    extern __shared__ float smem[];
    float2* z  = (float2*)smem;            // 32768 complex
    float*  sf = smem + 2 * FFT_NC;        // 8000 features
    __shared__ float ssum, ssum2;

    const int tid = threadIdx.x;
    const int row = blockIdx.x;
    const float* xr = x + (size_t)row * FFT_N;

    // ---- load row into LDS, natural order: z[n] = x[2n] + i*x[2n+1] -------
#if USE_TDM
    if (tid < 32) {                        // wave 0 issues one TDM descriptor
        const unsigned           lds_off = (unsigned)(uintptr_t)(void*)z;
        const unsigned long long ga      = (unsigned long long)(uintptr_t)xr;
        // D# group 0: count=1 | lds_addr | global_addr | type=2
        v4u g0;
        g0[0] = 1u;                                      // count=1, user D#
        g0[1] = lds_off;                                 // LDS byte address
        g0[2] = (unsigned)(ga & 0xFFFFFFFFu);            // global_addr lo
        g0[3] = (unsigned)((ga >> 32) & 0x01FFFFFFu)     // global_addr hi
                | 0x80000000u;                           // type = 2 ("image")
        // D# group 1: 2D tile 4096 x 16 of 4-byte elements, contiguous.
        v8i g1;
        g1[0] = 0x00020000;        // data_size=2 (4B); no multicast/pad/iter
        g1[1] = (4096 << 16);      // tensor_dim0[15:0]=4096 (hi bits 0)
        g1[2] = (16 << 16);        // tensor_dim0[31:16]=0 | tensor_dim1=16
        g1[3] = (4096 << 16);      // tensor_dim1 hi=0 | tile_dim0=4096
        g1[4] = 16;                // tile_dim1=16 | tile_dim2=0
        g1[5] = 4096;              // tensor_dim0_stride=4096 (elements)
        g1[6] = 0;                 // stride hi | tensor_dim1_stride lo
        g1[7] = 0;                 // tensor_dim1_stride hi
        v4i gz = {0, 0, 0, 0};
#if __has_include(<hip/amd_detail/amd_gfx1250_TDM.h>)
        v8i gz8 = {0, 0, 0, 0, 0, 0, 0, 0};
        __builtin_amdgcn_tensor_load_to_lds(g0, g1, gz, gz, gz8, 0);
#else
        __builtin_amdgcn_tensor_load_to_lds(g0, g1, gz, gz, 0);
#endif
        __builtin_amdgcn_s_wait_tensorcnt(0);
    }
    __syncthreads();
#else
    {
        const float2* xr2 = (const float2*)xr;
        for (int i = tid; i < FFT_NC; i += T1) z[i] = xr2[i];  // coalesced b64
        __syncthreads();
    }
#endif

    // ---- 15 radix-2 DIF stages in LDS -------------------------------------
    // For a fixed thread, pos = (tid + m*T1) & (half-1):
    //   half <= T1 : pos is constant -> twiddle hoisted (1 sincos/stage)
    //   half >  T1 : pos steps by T1 -> complex recurrence W^(pos+T1) =
    //                W^pos * W^T1, sign flip on wrap (W^half = -1).
    for (int s = 15; s >= 1; --s) {
        const int   half  = 1 << (s - 1);
        const float wstep = -PI_F / (float)half;
        float sn, cs, snd = 0.f, csd = 1.f;
        __sincosf(wstep * (float)(tid & (half - 1)), &sn, &cs);
        const bool stepped = (half > T1);
        if (stepped) __sincosf(wstep * (float)T1, &snd, &csd);

        for (int j = tid; j < (FFT_NC >> 1); j += T1) {
            const int pos = j & (half - 1);
            const int i0  = ((j >> (s - 1)) << s) + pos;
            const int i1  = i0 + half;
            const float2 a = z[i0];
            const float2 b = z[i1];
            z[i0] = make_float2(a.x + b.x, a.y + b.y);
            const float dr = a.x - b.x;
            const float di = a.y - b.y;
            z[i1] = make_float2(dr * cs - di * sn, dr * sn + di * cs);
            if (stepped) {                     // advance twiddle by W^T1
                float ncs = cs * csd - sn * snd;
                float nsn = cs * snd + sn * csd;
                if (pos + T1 >= half) { ncs = -ncs; nsn = -nsn; }  // wrap: *-1
                cs = ncs; sn = nsn;
            }
        }
        __syncthreads();
    }

    // ---- real-FFT untangle (bit-reversed indices), accumulate LN stats ----
    float lsum = 0.f, lsum2 = 0.f;
    for (int k = tid; k < KEEP; k += T1) {
        const int rk = (int)(__brev((unsigned)k) >> 17);
        const int rm = (int)(__brev((unsigned)((FFT_NC - k) & (FFT_NC - 1))) >> 17);
        const float2 Zk = z[rk];
        const float2 Zm = z[rm];
        // Even = 0.5*(Zk + conj(Zm)); Odd = -0.5i*(Zk - conj(Zm))
        const float er  = 0.5f * (Zk.x + Zm.x);
        const float ei  = 0.5f * (Zk.y - Zm.y);
        const float orr = 0.5f * (Zk.y + Zm.y);
        const float oi  = -0.5f * (Zk.x - Zm.x);
        float sn, cs;
        __sincosf(-TWO_PI_F * (float)k / (float)FFT_N, &sn, &cs);
        const float xre = er + cs * orr - sn * oi;
        const float xim = ei + cs * oi + sn * orr;
        sf[k]        = xre;
        sf[KEEP + k] = xim;
        lsum  += xre + xim;
        lsum2 += xre * xre + xim * xim;
    }
    if (tid == 0) { ssum = 0.f; ssum2 = 0.f; }
    __syncthreads();
    atomicAdd(&ssum, lsum);
    atomicAdd(&ssum2, lsum2);
    __syncthreads();

    const float mu  = ssum * (1.0f / FEAT);
    const float var = ssum2 * (1.0f / FEAT) - mu * mu;
    const float rs  = rsqrtf(var + 1e-5f);

    float* yr = y + (size_t)row * FEAT;
    for (int e = tid; e < FEAT; e += T1) {
        yr[e] = (sf[e] - mu) * rs * ln_w[e] + ln_b[e];
    }
}

// ---------------------------------------------------------------------------
// Kernel 2: out = silu(y @ W^T + b) with fp32 WMMA (V_WMMA_F32_16X16X4_F32).
// Block = 256 threads (8 waves). Block owns 16 rows of y (staged in LDS);
// each wave computes 4 column tiles of 16 (8 waves * 4 * 16 = 512 cols).
// ---------------------------------------------------------------------------
#define KCHUNK 64

__global__ __launch_bounds__(256)
void gemm_silu_kernel(const float* __restrict__ y,   /* BROWS x FEAT */
                      const float* __restrict__ W,   /* OUTC  x FEAT */
                      const float* __restrict__ bias,/* OUTC */
                      float* __restrict__ out)       /* BROWS x OUTC */
{
    __shared__ float As[16 * KCHUNK];   // 4 KB

    const int tid  = threadIdx.x;
    const int lane = tid & 31;
    const int wave = tid >> 5;
    const int m0   = blockIdx.x * 16;

    const int r     = lane & 15;   // row within tile (A) / col (B,D)
    const int hi    = lane >> 4;   // lane group 0/1
    const int khalf = hi * 2;      // K sub-offset: 0 or 2

    v8f acc[4];
    for (int t = 0; t < 4; ++t) acc[t] = (v8f){0.f,0.f,0.f,0.f,0.f,0.f,0.f,0.f};

    for (int k0 = 0; k0 < FEAT; k0 += KCHUNK) {
        __syncthreads();
        for (int i = tid; i < 16 * KCHUNK; i += 256) {
            const int rr = i >> 6;          // i / KCHUNK
            const int kk = i & (KCHUNK - 1);
            As[i] = y[(size_t)(m0 + rr) * FEAT + k0 + kk];
        }
        __syncthreads();

        for (int t = 0; t < 4; ++t) {
            const int ncol = (wave * 4 + t) * 16 + r;
            const float* Wn = W + (size_t)ncol * FEAT + k0;
            __builtin_prefetch(Wn + KCHUNK, 0, 1);   // global_prefetch_b8
            for (int ks = 0; ks < KCHUNK; ks += 4) {
                v2f a, b;
                // A 16x4 f32 fragment (ISA 7.12.2): lane r holds row M=r;
                // lanes 0-15 -> K = ks+{0,1}, lanes 16-31 -> K = ks+{2,3}
                a.x = As[r * KCHUNK + ks + khalf];
                a.y = As[r * KCHUNK + ks + khalf + 1];
                // B 4x16 fragment: lane r holds column N; K mirrors A layout.
                b.x = Wn[ks + khalf];
                b.y = Wn[ks + khalf + 1];
                acc[t] = __builtin_amdgcn_wmma_f32_16x16x4_f32(
                    false, a, false, b, (short)0, acc[t], false, false);
            }
        }
    }

    // epilogue: bias + SiLU. D layout: N = lane&15, M = v + 8*(lane>>4)
    for (int t = 0; t < 4; ++t) {
        const int ncol = (wave * 4 + t) * 16 + r;
        const float bv = bias[ncol];
        for (int v = 0; v < 8; ++v) {
            const int m = m0 + v + 8 * hi;
            const float val = acc[t][v] + bv;
            out[(size_t)m * OUTC + ncol] = val / (1.0f + __expf(-val));
        }
    }
}

// ---------------------------------------------------------------------------
extern "C" void kernel_launch(void* const* d_in, const int* in_sizes, int n_in,
                              void* d_out, int out_size, void* d_ws, size_t ws_size,
                              hipStream_t stream) {
    const float* x    = (const float*)d_in[0];   // 2048 x 65536
    const float* ln_w = (const float*)d_in[1];   // 8000
    const float* ln_b = (const float*)d_in[2];   // 8000
    const float* W    = (const float*)d_in[3];   // 512 x 8000
    const float* b    = (const float*)d_in[4];   // 512

    float* y   = (float*)d_ws;                   // 2048 x 8000 fp32 (65.5 MB)
    float* out = (float*)d_out;                  // 2048 x 512 fp32

    const size_t lds_bytes = (size_t)(2 * FFT_NC + FEAT) * sizeof(float); // ~295 KB
    hipFuncSetAttribute((const void*)fft_ln_kernel,
                        hipFuncAttributeMaxDynamicSharedMemorySize,
                        (int)lds_bytes);

    fft_ln_kernel<<<BROWS, T1, lds_bytes, stream>>>(x, ln_w, ln_b, y);
    gemm_silu_kernel<<<BROWS / 16, 256, 0, stream>>>(y, W, b, out);
}